// SheafGATConv_14336600834346
// MI455X (gfx1250) — compile-verified
//
#include <hip/hip_runtime.h>
#include <hip/hip_bf16.h>

#define IN_CH  128
#define OUT_CH 128
#define NEG_SLOPE 0.2f
#define EPS 1e-16f

typedef float v2f __attribute__((ext_vector_type(2)));
typedef float v8f __attribute__((ext_vector_type(8)));

// Order-preserving float->int encoding for integer atomicMax (exact, deterministic)
__device__ __forceinline__ int enc_f32(float f) {
    int i = __float_as_int(f);
    return (i < 0) ? (i ^ 0x7FFFFFFF) : i;
}
__device__ __forceinline__ float dec_f32(int i) {
    return __int_as_float((i < 0) ? (i ^ 0x7FFFFFFF) : i);
}

// ---------------------------------------------------------------------------
// Kernel 1: fused dense transforms via f32 WMMA (16x16x4).
//   mat in [0, n_types)  : h_all[n, mat, :] = x[n,:] @ W[mat]
//   mat == n_types       : out[n, :] = x[n,:] @ root_w^T + root_b
// One wave (32 threads) per 16x16 output tile; K=128 -> 32 WMMA steps.
// ---------------------------------------------------------------------------
__global__ void rgat_gemm_wmma(const float* __restrict__ x,
                               const float* __restrict__ weight,
                               const float* __restrict__ root_w,
                               const float* __restrict__ root_b,
                               float* __restrict__ h_all,
                               float* __restrict__ out,
                               int n_nodes, int n_types) {
    const int lane = threadIdx.x & 31;
    const int r    = lane & 15;     // row-in-half (A: M, B: N column)
    const int kh   = lane >> 4;     // half select (A: K pair 0/1, B: K-row pair)
    const int m0   = blockIdx.x * 16;
    const int g    = blockIdx.y;
    const int mat  = g >> 3;        // which weight matrix
    const int n0   = (g & 7) * 16;  // output column tile

    const int arow = min(m0 + r, n_nodes - 1);
    const float* aptr = x + (size_t)arow * IN_CH;

    v8f c = {0.f, 0.f, 0.f, 0.f, 0.f, 0.f, 0.f, 0.f};

    if (mat < n_types) {
        // B[k][n] = weight[mat][k][n]  (row-major [T, IN, OUT])
        const float* B = weight + (size_t)mat * IN_CH * OUT_CH;
        #pragma unroll
        for (int ks = 0; ks < IN_CH; ks += 4) {
            const int k = ks + 2 * kh;
            v2f a; a.x = aptr[k];            a.y = aptr[k + 1];
            v2f b; b.x = B[(size_t)k       * OUT_CH + n0 + r];
                   b.y = B[(size_t)(k + 1) * OUT_CH + n0 + r];
            c = __builtin_amdgcn_wmma_f32_16x16x4_f32(false, a, false, b,
                                                      (short)0, c, false, false);
        }
        #pragma unroll
        for (int i = 0; i < 8; ++i) {
            const int row = m0 + 8 * kh + i;
            if (row < n_nodes)
                h_all[((size_t)row * n_types + mat) * OUT_CH + n0 + r] = c[i];
        }
    } else {
        // B[k][n] = root_w[n][k]  (root_w is [OUT, IN] row-major -> contiguous f2)
        #pragma unroll
        for (int ks = 0; ks < IN_CH; ks += 4) {
            const int k = ks + 2 * kh;
            v2f a; a.x = aptr[k]; a.y = aptr[k + 1];
            const float* bp = root_w + (size_t)(n0 + r) * IN_CH + k;
            v2f b; b.x = bp[0]; b.y = bp[1];
            c = __builtin_amdgcn_wmma_f32_16x16x4_f32(false, a, false, b,
                                                      (short)0, c, false, false);
        }
        const float bias = root_b[n0 + r];
        #pragma unroll
        for (int i = 0; i < 8; ++i) {
            const int row = m0 + 8 * kh + i;
            if (row < n_nodes)
                out[(size_t)row * OUT_CH + n0 + r] = c[i] + bias;
        }
    }
}

// ---------------------------------------------------------------------------
// Kernel 2: init segment-max (encoded -inf floor) and denominators.
// ---------------------------------------------------------------------------
__global__ void rgat_init(int* __restrict__ seg_max, float* __restrict__ denom, int n) {
    const int i = blockIdx.x * blockDim.x + threadIdx.x;
    if (i < n) {
        seg_max[i] = (int)0x80000000;  // < enc(-inf)
        denom[i]   = 0.f;
    }
}

// ---------------------------------------------------------------------------
// Kernel 3: per-edge attention logits + segment max. One wave per edge,
// each lane handles 4 channels (float4), wave-wide shuffle reduction.
// ---------------------------------------------------------------------------
__global__ void rgat_logits(const float* __restrict__ h_all,
                            const int* __restrict__ edge_index,
                            const int* __restrict__ edge_type,
                            const float* __restrict__ att,
                            float* __restrict__ e_buf,
                            int* __restrict__ seg_max,
                            int n_edges, int n_types) {
    const int wid  = (blockIdx.x * blockDim.x + threadIdx.x) >> 5;
    const int lane = threadIdx.x & 31;
    if (wid >= n_edges) return;

    const int s = edge_index[wid];
    const int d = edge_index[n_edges + wid];
    const int t = edge_type[wid];

    const float4* hs = (const float4*)(h_all + ((size_t)s * n_types + t) * OUT_CH);
    const float4* hd = (const float4*)(h_all + ((size_t)d * n_types + t) * OUT_CH);
    const float4* al = (const float4*)(att + (size_t)t * 2 * OUT_CH);
    const float4* ar = (const float4*)(att + (size_t)t * 2 * OUT_CH + OUT_CH);

    const float4 a = hs[lane], b = hd[lane], u = al[lane], v = ar[lane];
    float p = a.x * u.x + a.y * u.y + a.z * u.z + a.w * u.w
            + b.x * v.x + b.y * v.y + b.z * v.z + b.w * v.w;

    p += __shfl_xor(p, 16, 32);
    p += __shfl_xor(p,  8, 32);
    p += __shfl_xor(p,  4, 32);
    p += __shfl_xor(p,  2, 32);
    p += __shfl_xor(p,  1, 32);

    if (lane == 0) {
        const float e = (p > 0.f) ? p : NEG_SLOPE * p;   // leaky relu
        e_buf[wid] = e;
        atomicMax(&seg_max[d], enc_f32(e));
    }
}

// ---------------------------------------------------------------------------
// Kernel 4: ex = exp(e - seg_max[dst]); denom[dst] += ex. One thread per edge.
// ---------------------------------------------------------------------------
__global__ void rgat_expsum(const float* __restrict__ e_buf,
                            const int* __restrict__ edge_index,
                            const int* __restrict__ seg_max,
                            float* __restrict__ ex_buf,
                            float* __restrict__ denom,
                            int n_edges) {
    const int e = blockIdx.x * blockDim.x + threadIdx.x;
    if (e >= n_edges) return;
    const int d = edge_index[n_edges + e];
    const float m  = dec_f32(seg_max[d]);
    const float ex = __expf(e_buf[e] - m);
    ex_buf[e] = ex;
    atomicAdd(&denom[d], ex);
}

// ---------------------------------------------------------------------------
// Kernel 5: out[dst,:] += alpha * h_src. One wave per edge, lane = 4 channels.
// Gathers are L2-resident (h_all is 51.2 MB < 192 MB L2).
// ---------------------------------------------------------------------------
__global__ void rgat_scatter(const float* __restrict__ h_all,
                             const int* __restrict__ edge_index,
                             const int* __restrict__ edge_type,
                             const float* __restrict__ ex_buf,
                             const float* __restrict__ denom,
                             float* __restrict__ out,
                             int n_edges, int n_types) {
    const int wid  = (blockIdx.x * blockDim.x + threadIdx.x) >> 5;
    const int lane = threadIdx.x & 31;
    if (wid >= n_edges) return;

    const int s = edge_index[wid];
    const int d = edge_index[n_edges + wid];
    const int t = edge_type[wid];

    const float alpha = ex_buf[wid] / (denom[d] + EPS);

    const float4* hs = (const float4*)(h_all + ((size_t)s * n_types + t) * OUT_CH);
    const float4 h = hs[lane];

    float* o = out + (size_t)d * OUT_CH + lane * 4;
    atomicAdd(o + 0, h.x * alpha);
    atomicAdd(o + 1, h.y * alpha);
    atomicAdd(o + 2, h.z * alpha);
    atomicAdd(o + 3, h.w * alpha);
}

// ---------------------------------------------------------------------------
extern "C" void kernel_launch(void* const* d_in, const int* in_sizes, int n_in,
                              void* d_out, int out_size, void* d_ws, size_t ws_size,
                              hipStream_t stream) {
    const float* x      = (const float*)d_in[0];
    const int*   ei     = (const int*)  d_in[1];   // [2, E]
    const int*   et     = (const int*)  d_in[2];   // [E]
    const float* weight = (const float*)d_in[3];   // [T, IN, OUT]
    const float* att    = (const float*)d_in[4];   // [T, 2*OUT]
    const float* root_w = (const float*)d_in[5];   // [OUT, IN]
    const float* root_b = (const float*)d_in[6];   // [OUT]
    float* out = (float*)d_out;

    const int N = in_sizes[0] / IN_CH;
    const int E = in_sizes[2];
    const int T = in_sizes[4] / (2 * OUT_CH);

    // workspace layout
    float* h_all   = (float*)d_ws;
    float* e_buf   = h_all + (size_t)N * T * OUT_CH;
    float* ex_buf  = e_buf + E;
    float* denom   = ex_buf + E;
    int*   seg_max = (int*)(denom + N);

    // 1) dense transforms (WMMA): h_all and out = x@root_w^T + b
    {
        dim3 grid((N + 15) / 16, (T + 1) * 8);
        rgat_gemm_wmma<<<grid, 32, 0, stream>>>(x, weight, root_w, root_b,
                                                h_all, out, N, T);
    }
    // 2) init segment stats
    rgat_init<<<(N + 255) / 256, 256, 0, stream>>>(seg_max, denom, N);
    // 3) logits + segment max (wave per edge)
    {
        const int waves_per_block = 256 / 32;
        dim3 grid((E + waves_per_block - 1) / waves_per_block);
        rgat_logits<<<grid, 256, 0, stream>>>(h_all, ei, et, att, e_buf, seg_max, E, T);
    }
    // 4) exp + denominator
    rgat_expsum<<<(E + 255) / 256, 256, 0, stream>>>(e_buf, ei, seg_max, ex_buf, denom, E);
    // 5) weighted scatter-add (wave per edge)
    {
        const int waves_per_block = 256 / 32;
        dim3 grid((E + waves_per_block - 1) / waves_per_block);
        rgat_scatter<<<grid, 256, 0, stream>>>(h_all, ei, et, ex_buf, denom, out, E, T);
    }
}